// genpatchwithMaskEntropy_80788334837908
// MI455X (gfx1250) — compile-verified
//
#include <hip/hip_runtime.h>

// genpatchwithMaskEntropy for MI455X (gfx1250, wave32).
// Pipeline: softmax/entropy maps -> WMMA banded-GEMM sliding window sums
// (horizontal, then vertical) -> greedy argmax + box suppression -> patch gather.

typedef __attribute__((ext_vector_type(2))) float v2f;
typedef __attribute__((ext_vector_type(8))) float v8f;

#define BATCH 32
#define HW    256
#define POOL  225     // 256 - 32 + 1
#define EPSF  1e-5f

// ---------------------------------------------------------------------------
// Kernel 1: per-pixel softmax prob0 and summed channel entropy.
// maps[b][0][y][x] = p0, maps[b][1][y][x] = -(p0*log(p0+eps)+p1*log(p1+eps))
// ---------------------------------------------------------------------------
__global__ void map_kernel(const float* __restrict__ infeat, float* __restrict__ maps) {
  int idx = blockIdx.x * blockDim.x + threadIdx.x;
  if (idx >= BATCH * HW * HW) return;
  int b   = idx >> 16;
  int pix = idx & 65535;
  float a0 = infeat[(size_t)(b * 2 + 0) * 65536 + pix];
  float a1 = infeat[(size_t)(b * 2 + 1) * 65536 + pix];
  float m  = fmaxf(a0, a1);
  float e0 = __expf(a0 - m), e1 = __expf(a1 - m);
  float inv = 1.0f / (e0 + e1);
  float p0 = e0 * inv, p1 = e1 * inv;
  float ent = -(p0 * __logf(p0 + EPSF) + p1 * __logf(p1 + EPSF));
  maps[(size_t)(b * 2 + 0) * 65536 + pix] = p0;
  maps[(size_t)(b * 2 + 1) * 65536 + pix] = ent;
}

// ---------------------------------------------------------------------------
// Banded ones matrix S[k][n] = (n <= k < n+32), k in [0,48), n in [0,16).
// Sliding 32-wide window sum as D = A(16x48) x S(48x16) via 12 chained
// V_WMMA_F32_16X16X4_F32, band generated in registers.
// ---------------------------------------------------------------------------

// Horizontal pass: maps [16384 rows][256] -> mid [16384 rows][225]
__global__ void hpool_wmma(const float* __restrict__ maps, float* __restrict__ mid) {
  int wid = blockIdx.x * (blockDim.x >> 5) + (threadIdx.x >> 5);   // wave id = tile id
  const int nTiles = 1024 * 15;                                    // 16384/16 row tiles x ceil(225/16)
  if (wid >= nTiles) return;                                       // wave-uniform (EXEC stays full)
  int rt = wid / 15, ct = wid % 15;
  int lane = threadIdx.x & 31;
  int half = lane >> 4;          // K sub-block select
  int ln   = lane & 15;          // A: row M ; B/D: col N
  int row  = rt * 16 + ln;
  int j0   = ct * 16;
  const float* src = maps + (size_t)row * 256;
  v8f acc = {};
  #pragma unroll
  for (int kb = 0; kb < 48; kb += 4) {
    int k0 = kb + (half ? 2 : 0);
    int k1 = k0 + 1;
    int i0 = j0 + k0; if (i0 > 255) i0 = 255;   // clamp: band entry is 0 there
    int i1 = j0 + k1; if (i1 > 255) i1 = 255;
    v2f a, bmat;
    a.x = src[i0];
    a.y = src[i1];
    bmat.x = (k0 >= ln && k0 < ln + 32) ? 1.0f : 0.0f;
    bmat.y = (k1 >= ln && k1 < ln + 32) ? 1.0f : 0.0f;
    acc = __builtin_amdgcn_wmma_f32_16x16x4_f32(false, a, false, bmat,
                                                (short)0, acc, false, false);
  }
  int outcol = j0 + ln;                                // N index
  if (outcol < POOL) {
    #pragma unroll
    for (int r = 0; r < 8; ++r) {                      // D: VGPR r -> M = r (+8 upper half)
      int M = r + (half ? 8 : 0);
      mid[(size_t)(rt * 16 + M) * POOL + outcol] = acc[r];
    }
  }
}

// Vertical pass: mid [64 maps][256][225] -> pooled [64 maps][225][225] (/1024)
// Here M = x column, N = output y, K = input y.
__global__ void vpool_wmma(const float* __restrict__ mid, float* __restrict__ pooled) {
  int wid = blockIdx.x * (blockDim.x >> 5) + (threadIdx.x >> 5);
  const int nTiles = 64 * 15 * 15;
  if (wid >= nTiles) return;
  int bm = wid / 225;
  int t  = wid % 225;
  int yt = t / 15, xt = t % 15;
  int lane = threadIdx.x & 31;
  int half = lane >> 4;
  int ln   = lane & 15;
  int y0 = yt * 16, x0 = xt * 16;
  int xm = x0 + ln; if (xm > 224) xm = 224;            // clamp (invalid M discarded at store)
  const float* src = mid + (size_t)bm * (256 * POOL);
  v8f acc = {};
  #pragma unroll
  for (int kb = 0; kb < 48; kb += 4) {
    int k0 = kb + (half ? 2 : 0);
    int k1 = k0 + 1;
    int yA = y0 + k0; if (yA > 255) yA = 255;
    int yB = y0 + k1; if (yB > 255) yB = 255;
    v2f a, bmat;
    a.x = src[(size_t)yA * POOL + xm];
    a.y = src[(size_t)yB * POOL + xm];
    bmat.x = (k0 >= ln && k0 < ln + 32) ? 1.0f : 0.0f;
    bmat.y = (k1 >= ln && k1 < ln + 32) ? 1.0f : 0.0f;
    acc = __builtin_amdgcn_wmma_f32_16x16x4_f32(false, a, false, bmat,
                                                (short)0, acc, false, false);
  }
  float* dst = pooled + (size_t)bm * (POOL * POOL);
  int yj = y0 + ln;                                    // N index
  if (yj < POOL) {
    #pragma unroll
    for (int r = 0; r < 8; ++r) {
      int M = r + (half ? 8 : 0);
      int x = x0 + M;
      if (x < POOL) dst[(size_t)yj * POOL + x] = acc[r] * (1.0f / 1024.0f);
    }
  }
}

// ---------------------------------------------------------------------------
// Kernel 4: greedy top-2 with box suppression per (b, c). 64 blocks.
// filt = (c==0 ? featP0 : 1-featP0) - 0.1*entP ; suppression multiplies by 0.
// ---------------------------------------------------------------------------
__global__ void select_kernel(const float* __restrict__ pooled,
                              float* __restrict__ out_val,
                              float* __restrict__ out_xy,
                              int* __restrict__ selinfo) {
  int b = blockIdx.x >> 1;
  int c = blockIdx.x & 1;
  const float* p = pooled + (size_t)(b * 2 + 0) * (POOL * POOL);
  const float* e = pooled + (size_t)(b * 2 + 1) * (POOL * POOL);
  __shared__ float svals[256];
  __shared__ int   sidx[256];
  __shared__ int   sbox[4];
  int tid = threadIdx.x;
  for (int kk = 0; kk < 2; ++kk) {
    float bv = -1e30f; int bi = 0x7fffffff;
    for (int idx = tid; idx < POOL * POOL; idx += 256) {
      int y = idx / POOL, x = idx % POOL;
      float fp = p[idx];
      float v = ((c == 0) ? fp : (1.0f - fp)) - 0.1f * e[idx];
      if (kk == 1 && y >= sbox[0] && y < sbox[1] && x >= sbox[2] && x < sbox[3])
        v = 0.0f;                                     // mask * filt
      if (v > bv || (v == bv && idx < bi)) { bv = v; bi = idx; }
    }
    svals[tid] = bv; sidx[tid] = bi;
    __syncthreads();
    for (int s = 128; s > 0; s >>= 1) {
      if (tid < s) {
        float ov = svals[tid + s]; int oi = sidx[tid + s];
        if (ov > svals[tid] || (ov == svals[tid] && oi < sidx[tid])) {
          svals[tid] = ov; sidx[tid] = oi;
        }
      }
      __syncthreads();
    }
    if (tid == 0) {
      int idx = sidx[0];
      int py = idx / POOL, px = idx % POOL;
      int n = (c * 2 + kk) * BATCH + b;               // concat order: (c, kk, b)
      out_val[n] = svals[0];
      out_xy[n * 4 + 0] = (float)px;                  // pointXY[n] = [[px,px+31],[py,py+31]]
      out_xy[n * 4 + 1] = (float)(px + 31);
      out_xy[n * 4 + 2] = (float)py;
      out_xy[n * 4 + 3] = (float)(py + 31);
      selinfo[n * 3 + 0] = b;
      selinfo[n * 3 + 1] = py;
      selinfo[n * 3 + 2] = px;
      sbox[0] = py - 16 < 0 ? 0 : py - 16;
      sbox[1] = py + 16 > POOL ? POOL : py + 16;
      sbox[2] = px - 16 < 0 ? 0 : px - 16;
      sbox[3] = px + 16 > POOL ? POOL : px + 16;
    }
    __syncthreads();
  }
}

// ---------------------------------------------------------------------------
// Kernel 5: coalesced 32x32 patch gather. block = (n*C + ch), 128 threads.
// ---------------------------------------------------------------------------
__global__ void patch_copy(const float* __restrict__ src, int C,
                           float* __restrict__ dst, const int* __restrict__ selinfo) {
  int blk = blockIdx.x;
  int n = blk / C, ch = blk % C;
  int b  = selinfo[n * 3 + 0];
  int py = selinfo[n * 3 + 1];
  int px = selinfo[n * 3 + 2];
  const float* s = src + (size_t)(b * C + ch) * 65536;
  float* d = dst + (size_t)blk * 1024;
  int tid = threadIdx.x;
  #pragma unroll
  for (int it = 0; it < 8; ++it) {
    int el  = it * 128 + tid;
    int r   = el >> 5;
    int col = el & 31;
    d[el] = s[(size_t)(py + r) * 256 + (px + col)];
  }
}

// ---------------------------------------------------------------------------
extern "C" void kernel_launch(void* const* d_in, const int* in_sizes, int n_in,
                              void* d_out, int out_size, void* d_ws, size_t ws_size,
                              hipStream_t stream) {
  const float* infeat       = (const float*)d_in[0];
  const float* labelTpesudo = (const float*)d_in[1];
  const float* labelT       = (const float*)d_in[2];
  const float* FeatureDA    = (const float*)d_in[3];
  float* out = (float*)d_out;
  float* ws  = (float*)d_ws;

  // workspace layout (floats):
  //   [0, 4194304)           maps  [32][2][256][256]
  //   [4194304, 7880704)     mid   [32*2][256][225]
  //   pooled [32*2][225][225] (3240000) aliases maps (maps dead after hpool)
  float* maps   = ws;
  float* mid    = ws + 4194304;
  float* pooled = ws;                         // alias, safe: vpool reads mid only
  int*   selinfo = (int*)(ws + 4194304 + 3686400);

  // output offsets (floats)
  float* out_cls  = out;                      // [128][2][32][32]
  float* out_feat = out + 262144;             // [128][64][32][32]
  float* out_lp   = out + 8650752;            // [128][1][32][32]
  float* out_lt   = out + 8781824;            // [128][1][32][32]
  float* out_val  = out + 8912896;            // [128]
  float* out_xy   = out + 8913024;            // [128][2][2]

  map_kernel <<<(BATCH * HW * HW + 255) / 256, 256, 0, stream>>>(infeat, maps);
  hpool_wmma <<<(1024 * 15) / 4,      128, 0, stream>>>(maps, mid);     // 4 waves/block
  vpool_wmma <<<(64 * 15 * 15) / 4,   128, 0, stream>>>(mid, pooled);
  select_kernel<<<64, 256, 0, stream>>>(pooled, out_val, out_xy, selinfo);
  patch_copy <<<128 * 2,  128, 0, stream>>>(infeat,       2,  out_cls,  selinfo);
  patch_copy <<<128 * 64, 128, 0, stream>>>(FeatureDA,    64, out_feat, selinfo);
  patch_copy <<<128 * 1,  128, 0, stream>>>(labelTpesudo, 1,  out_lp,   selinfo);
  patch_copy <<<128 * 1,  128, 0, stream>>>(labelT,       1,  out_lt,   selinfo);
}